// DecoderWithAttention_23098334118093
// MI455X (gfx1250) — compile-verified
//
#include <hip/hip_runtime.h>
#include <hip/hip_bf16.h>
#include <math.h>

// Problem constants (from reference)
#define BB 128
#define PP 196
#define ENCD 2048
#define AD 512
#define ED 512
#define DD 512
#define VD 10000
#define LD 52
#define TT 51   // L - 1

typedef __attribute__((ext_vector_type(2))) float v2f;
typedef __attribute__((ext_vector_type(8))) float v8f;

__device__ __forceinline__ float sigf(float x) { return 1.0f / (1.0f + __expf(-x)); }

#define WMMA_F32(a, b, c) \
    __builtin_amdgcn_wmma_f32_16x16x4_f32(false, (a), false, (b), (short)0, (c), false, false)

// ---------------------------------------------------------------------------
// Setup: stable descending argsort of lengths, dec_len, sorted caps -> d_out,
// zero-init h and c. Single block of 128 threads.
// ---------------------------------------------------------------------------
__global__ void setup_kernel(const int* __restrict__ clens,   // (B,1)
                             const int* __restrict__ caps,    // (B,L)
                             int* __restrict__ order,         // (B)
                             int* __restrict__ declen,        // (B)
                             float* __restrict__ out_caps,    // (B,L) in d_out
                             float* __restrict__ out_declen,  // (B)   in d_out
                             float* __restrict__ h, float* __restrict__ c)
{
    __shared__ int slen[BB];
    int b = threadIdx.x;
    int mylen = clens[b];
    slen[b] = mylen;
    __syncthreads();
    // stable descending rank
    int rank = 0;
    for (int j = 0; j < BB; ++j) {
        int lj = slen[j];
        if (lj > mylen || (lj == mylen && j < b)) rank++;
    }
    order[rank]  = b;
    declen[rank] = mylen - 1;
    out_declen[rank] = (float)(mylen - 1);
    for (int i = 0; i < LD; ++i)
        out_caps[(size_t)rank * LD + i] = (float)caps[(size_t)b * LD + i];
    // zero h, c
    for (int i = b; i < BB * DD; i += BB) { h[i] = 0.0f; c[i] = 0.0f; }
}

// ---------------------------------------------------------------------------
// WMMA fragment layouts per CDNA5 ISA 7.12.2 (wave32):
//   A 16x4 : lanes 0-15 -> M=lane, K={0,1}; lanes 16-31 -> M=lane-16, K={2,3}
//   B 4x16 : lanes 0-15 -> N=lane, K={0,1}; lanes 16-31 -> N=lane-16, K={2,3}
//   C 16x16: VGPR r -> M = r + 8*half, N = lane&15
// ---------------------------------------------------------------------------
__device__ __forceinline__ void store_tile(float* __restrict__ C, long ldc,
                                           int mbase, int n, const v8f& acc,
                                           float bs, int hk,
                                           const int* __restrict__ declen, int t)
{
#pragma unroll
    for (int r = 0; r < 8; ++r) {
        int m = mbase + r + hk * 8;
        long off = (long)m * ldc + n;
        float v = acc[r] + bs;
        if (declen && t >= declen[m]) v = 0.0f;
        C[off] = v;
    }
}

// 2 (M) x NC (N) tile inner loop: 2+NC loads -> 2*NC WMMAs per K-step.
template<int NC>
__device__ __forceinline__ void mm_block2xN(const float* __restrict__ a0p,
                                            const float* __restrict__ a1p,
                                            const float* const* bp, int K,
                                            v8f* acc0, v8f* acc1)
{
    __builtin_prefetch(bp[0], 0, 1);   // global_prefetch_b8
    for (int k = 0; k < K; k += 4) {
        v2f a0 = *(const v2f*)(a0p + k);
        v2f a1 = *(const v2f*)(a1p + k);
#pragma unroll
        for (int j = 0; j < NC; ++j) {
            v2f b = *(const v2f*)(bp[j] + k);
            acc0[j] = WMMA_F32(a0, b, acc0[j]);
            acc1[j] = WMMA_F32(a1, b, acc1[j]);
        }
    }
}

// ---------------------------------------------------------------------------
// Generic f32 WMMA GEMM with 2x4 register blocking:
//   C[m,n] = sum_k A[m,k]*Bt[n,k] (+ sum_k2 A2[m,k]*Bt2[n,k]) + bias[n] (+bias2[n])
// One wave per 32x64 output block (2x4 tiles of 16x16). M % 32 == 0 required.
// N-tile remainder handled by templated column count (uniform per wave).
// declen/t: epilogue mask (write 0 where t >= declen[m]) for preds.
// ---------------------------------------------------------------------------
__global__ void gemm_wmma_f32(const float* __restrict__ A, int lda,
                              const float* __restrict__ Bt, int ldb,
                              const float* __restrict__ bias,
                              const float* __restrict__ A2, int lda2,
                              const float* __restrict__ Bt2, int ldb2, int K2,
                              const float* __restrict__ bias2,
                              float* __restrict__ C, long ldc,
                              int M, int N, int K,
                              const int* __restrict__ declen, int t)
{
    const int wave = blockIdx.x * (blockDim.x >> 5) + (threadIdx.x >> 5);
    const int nt  = N >> 4;
    const int mtb = M >> 5;             // 2 M-tiles per wave
    const int ntb = (nt + 3) >> 2;      // 4 N-tiles per wave (last may be partial)
    if (wave >= mtb * ntb) return;      // uniform per wave: EXEC stays all-1s
    const int tm0 = (wave % mtb) * 2;
    const int tn0 = (wave / mtb) * 4;
    const int nc  = (nt - tn0) < 4 ? (nt - tn0) : 4;

    const int lane = threadIdx.x & 31;
    const int hk = lane >> 4;           // which K-pair this half-wave owns
    const int lq = lane & 15;

    v8f acc0[4] = {{}, {}, {}, {}};
    v8f acc1[4] = {{}, {}, {}, {}};

    {
        const float* a0p = A + (long)(tm0 * 16 + lq) * lda + hk * 2;
        const float* a1p = a0p + (long)16 * lda;
        const float* bp[4];
#pragma unroll
        for (int j = 0; j < 4; ++j) {
            int tn = (j < nc) ? (tn0 + j) : tn0;
            bp[j] = Bt + (long)(tn * 16 + lq) * ldb + hk * 2;
        }
        switch (nc) {
        case 4: mm_block2xN<4>(a0p, a1p, bp, K, acc0, acc1); break;
        case 3: mm_block2xN<3>(a0p, a1p, bp, K, acc0, acc1); break;
        case 2: mm_block2xN<2>(a0p, a1p, bp, K, acc0, acc1); break;
        default: mm_block2xN<1>(a0p, a1p, bp, K, acc0, acc1); break;
        }
    }
    if (A2) {   // fused second accumulation (gates: + h @ W_hh^T)
        const float* a0p = A2 + (long)(tm0 * 16 + lq) * lda2 + hk * 2;
        const float* a1p = a0p + (long)16 * lda2;
        const float* bp[4];
#pragma unroll
        for (int j = 0; j < 4; ++j) {
            int tn = (j < nc) ? (tn0 + j) : tn0;
            bp[j] = Bt2 + (long)(tn * 16 + lq) * ldb2 + hk * 2;
        }
        switch (nc) {
        case 4: mm_block2xN<4>(a0p, a1p, bp, K2, acc0, acc1); break;
        case 3: mm_block2xN<3>(a0p, a1p, bp, K2, acc0, acc1); break;
        case 2: mm_block2xN<2>(a0p, a1p, bp, K2, acc0, acc1); break;
        default: mm_block2xN<1>(a0p, a1p, bp, K2, acc0, acc1); break;
        }
    }

#pragma unroll
    for (int j = 0; j < 4; ++j) {
        if (j < nc) {
            int n = (tn0 + j) * 16 + lq;
            float bs = (bias ? bias[n] : 0.0f) + (bias2 ? bias2[n] : 0.0f);
            store_tile(C, ldc, tm0 * 16,      n, acc0[j], bs, hk, declen, t);
            store_tile(C, ldc, tm0 * 16 + 16, n, acc1[j], bs, hk, declen, t);
        }
    }
}

// ---------------------------------------------------------------------------
// enc_att[b,p,a] = sum_e enc[order[b],p,e] * We[a,e] + be[a]
// 4x4-blocked WMMA (8 loads -> 16 WMMAs per K-step); A rows gathered through
// the batch permutation. M = 25088 (1568 tiles, /4 = 392), N = 512 (32, /4 = 8).
// ---------------------------------------------------------------------------
__global__ void enc_att_wmma(const float* __restrict__ enc,
                             const int* __restrict__ order,
                             const float* __restrict__ We,
                             const float* __restrict__ be,
                             float* __restrict__ out)
{
    const int wave = blockIdx.x * (blockDim.x >> 5) + (threadIdx.x >> 5);
    const int mtb = (BB * PP) >> 6;     // 392
    const int ntb = AD >> 6;            // 8
    if (wave >= mtb * ntb) return;
    const int tm0 = (wave % mtb) * 4;
    const int tn0 = (wave / mtb) * 4;

    const int lane = threadIdx.x & 31;
    const int hk = lane >> 4, lq = lane & 15;

    const float* ap[4];
    const float* bp[4];
#pragma unroll
    for (int i = 0; i < 4; ++i) {
        int m = (tm0 + i) * 16 + lq;
        int g = order[m / PP] * PP + (m % PP);
        ap[i] = enc + (long)g * ENCD + hk * 2;
        int n = (tn0 + i) * 16 + lq;
        bp[i] = We + (long)n * ENCD + hk * 2;
    }
    __builtin_prefetch(bp[0], 0, 1);   // global_prefetch_b8

    v8f acc[4][4];
#pragma unroll
    for (int i = 0; i < 4; ++i)
#pragma unroll
        for (int j = 0; j < 4; ++j) acc[i][j] = (v8f){};

#pragma unroll 2
    for (int k = 0; k < ENCD; k += 4) {
        v2f a[4], b[4];
#pragma unroll
        for (int i = 0; i < 4; ++i) {
            a[i] = *(const v2f*)(ap[i] + k);
            b[i] = *(const v2f*)(bp[i] + k);
        }
#pragma unroll
        for (int i = 0; i < 4; ++i)
#pragma unroll
            for (int j = 0; j < 4; ++j)
                acc[i][j] = WMMA_F32(a[i], b[j], acc[i][j]);
    }

#pragma unroll
    for (int j = 0; j < 4; ++j) {
        int n = (tn0 + j) * 16 + lq;
        float bs = be[n];
#pragma unroll
        for (int i = 0; i < 4; ++i)
            store_tile(out, AD, (tm0 + i) * 16, n, acc[i][j], bs, hk, nullptr, 0);
    }
}

// ---------------------------------------------------------------------------
// scores[b,p] = bf + sum_a wf[a] * relu(enc_att[b,p,a] + dec_att[b,a])
// One wave per (b,p); lane-strided reduce over A=512.
// ---------------------------------------------------------------------------
__global__ void scores_kernel(const float* __restrict__ enc_att,
                              const float* __restrict__ dec_att,
                              const float* __restrict__ wf,
                              const float* __restrict__ bf,
                              float* __restrict__ scores)
{
    const int wid = blockIdx.x * (blockDim.x >> 5) + (threadIdx.x >> 5);
    if (wid >= BB * PP) return;
    const int b = wid / PP;
    const int lane = threadIdx.x & 31;
    const float* ea = enc_att + (long)wid * AD;
    const float* da = dec_att + (long)b * AD;
    float s = 0.0f;
#pragma unroll 4
    for (int a = lane; a < AD; a += 32) {
        float v = ea[a] + da[a];
        v = v > 0.0f ? v : 0.0f;
        s += wf[a] * v;
    }
#pragma unroll
    for (int off = 16; off > 0; off >>= 1) s += __shfl_xor(s, off, 32);
    if (lane == 0) scores[wid] = s + bf[0];
}

// ---------------------------------------------------------------------------
// Softmax over P=196, one block per b.
// ---------------------------------------------------------------------------
__global__ void softmax_kernel(const float* __restrict__ scores,
                               float* __restrict__ alpha)
{
    __shared__ float red[256];
    const int b = blockIdx.x, tid = threadIdx.x;
    float v = (tid < PP) ? scores[(long)b * PP + tid] : -INFINITY;
    red[tid] = v;
    __syncthreads();
    for (int s = 128; s > 0; s >>= 1) {
        if (tid < s) red[tid] = fmaxf(red[tid], red[tid + s]);
        __syncthreads();
    }
    const float mx = red[0];
    __syncthreads();
    float e = (tid < PP) ? __expf(v - mx) : 0.0f;
    red[tid] = e;
    __syncthreads();
    for (int s = 128; s > 0; s >>= 1) {
        if (tid < s) red[tid] += red[tid + s];
        __syncthreads();
    }
    const float inv = 1.0f / red[0];
    if (tid < PP) alpha[(long)b * PP + tid] = e * inv;
}

// ---------------------------------------------------------------------------
// awe[b,e] = sum_p alpha[b,p] * enc[order[b],p,e]   (ungated)
// grid (B, ENC/256); coalesced across e per p; 4 independent partials for MLP.
// ---------------------------------------------------------------------------
__global__ void awe_kernel(const float* __restrict__ alpha,
                           const float* __restrict__ enc,
                           const int* __restrict__ order,
                           float* __restrict__ awe)
{
    const int b = blockIdx.x;
    const int e = blockIdx.y * blockDim.x + threadIdx.x;
    const float* erow = enc + (long)order[b] * PP * ENCD + e;
    const float* arow = alpha + (long)b * PP;
    float s0 = 0.0f, s1 = 0.0f, s2 = 0.0f, s3 = 0.0f;
    for (int p = 0; p + 4 <= PP; p += 4) {
        s0 += arow[p + 0] * erow[(long)(p + 0) * ENCD];
        s1 += arow[p + 1] * erow[(long)(p + 1) * ENCD];
        s2 += arow[p + 2] * erow[(long)(p + 2) * ENCD];
        s3 += arow[p + 3] * erow[(long)(p + 3) * ENCD];
    }
    // PP = 196 = 4*49: no tail
    awe[(long)b * ENCD + e] = (s0 + s1) + (s2 + s3);
}

// ---------------------------------------------------------------------------
// x[b, 0:E]       = emb[caps_sorted[b][t]]
// x[b, E:E+ENC]   = sigmoid(gatepre[b,e]) * awe[b,e]
// ---------------------------------------------------------------------------
__global__ void xassemble_kernel(const float* __restrict__ emb,
                                 const int* __restrict__ caps,
                                 const int* __restrict__ order,
                                 const float* __restrict__ gatepre,
                                 const float* __restrict__ awe,
                                 float* __restrict__ x, int t)
{
    const long idx = (long)blockIdx.x * blockDim.x + threadIdx.x;
    if (idx >= (long)BB * (ED + ENCD)) return;
    const int b = (int)(idx / (ED + ENCD));
    const int j = (int)(idx % (ED + ENCD));
    float v;
    if (j < ED) {
        int tok = caps[(long)order[b] * LD + t];
        v = emb[(long)tok * ED + j];
    } else {
        int e = j - ED;
        long o = (long)b * ENCD + e;
        v = sigf(gatepre[o]) * awe[o];
    }
    x[idx] = v;
}

// ---------------------------------------------------------------------------
// LSTM cell; masked update of h,c; h_new kept unmasked for preds.
// gates layout: [i | f | g | o], each D wide.
// ---------------------------------------------------------------------------
__global__ void lstm_kernel(const float* __restrict__ gates,
                            float* __restrict__ h, float* __restrict__ c,
                            float* __restrict__ hnew,
                            const int* __restrict__ declen, int t)
{
    const int idx = blockIdx.x * blockDim.x + threadIdx.x;
    if (idx >= BB * DD) return;
    const int b = idx / DD, j = idx % DD;
    const float* g = gates + (long)b * (4 * DD);
    float gi = g[j], gf = g[DD + j], gg = g[2 * DD + j], go = g[3 * DD + j];
    float c_old = c[idx];
    float cn = sigf(gf) * c_old + sigf(gi) * tanhf(gg);
    float hn = sigf(go) * tanhf(cn);
    bool act = t < declen[b];
    hnew[idx] = hn;
    h[idx] = act ? hn : h[idx];
    c[idx] = act ? cn : c_old;
}

// ---------------------------------------------------------------------------
// Host-side launch
// ---------------------------------------------------------------------------
extern "C" void kernel_launch(void* const* d_in, const int* in_sizes, int n_in,
                              void* d_out, int out_size, void* d_ws, size_t ws_size,
                              hipStream_t stream)
{
    const float* enc   = (const float*)d_in[0];
    const int*   caps  = (const int*)  d_in[1];
    const int*   clens = (const int*)  d_in[2];
    const float* emb   = (const float*)d_in[3];
    const float* We    = (const float*)d_in[4];
    const float* be    = (const float*)d_in[5];
    const float* Wd    = (const float*)d_in[6];
    const float* bd    = (const float*)d_in[7];
    const float* wf    = (const float*)d_in[8];
    const float* bf    = (const float*)d_in[9];
    const float* W_ih  = (const float*)d_in[10];
    const float* b_ih  = (const float*)d_in[11];
    const float* W_hh  = (const float*)d_in[12];
    const float* b_hh  = (const float*)d_in[13];
    const float* Wbeta = (const float*)d_in[14];
    const float* bbeta = (const float*)d_in[15];
    const float* Wfc   = (const float*)d_in[16];
    const float* bfc   = (const float*)d_in[17];

    float* out = (float*)d_out;
    float* out_preds  = out;                                  // (B, T, V)
    float* out_caps   = out + (size_t)BB * TT * VD;           // (B, L)
    float* out_declen = out_caps + (size_t)BB * LD;           // (B)

    // workspace layout (floats)
    float* ws = (float*)d_ws;
    size_t o = 0;
    float* enc_att = ws + o; o += (size_t)BB * PP * AD;   // 12.8M
    float* h       = ws + o; o += (size_t)BB * DD;
    float* cbuf    = ws + o; o += (size_t)BB * DD;
    float* hnew    = ws + o; o += (size_t)BB * DD;
    float* dec_att = ws + o; o += (size_t)BB * AD;
    float* scores  = ws + o; o += (size_t)BB * PP;
    float* alpha   = ws + o; o += (size_t)BB * PP;
    float* awe     = ws + o; o += (size_t)BB * ENCD;
    float* gatepre = ws + o; o += (size_t)BB * ENCD;
    float* xbuf    = ws + o; o += (size_t)BB * (ED + ENCD);
    float* gates   = ws + o; o += (size_t)BB * 4 * DD;
    int*   order   = (int*)(ws + o); o += BB;
    int*   declen  = (int*)(ws + o); o += BB;

    // 1) sort batch, emit caps/dec_len outputs, zero h/c
    setup_kernel<<<1, BB, 0, stream>>>(clens, caps, order, declen,
                                       out_caps, out_declen, h, cbuf);

    // 2) enc_att = einsum('bpe,ae->bpa') + be : 392*8 = 3136 waves (4x4 tiles)
    {
        int waves = ((BB * PP) >> 6) * (AD >> 6);
        enc_att_wmma<<<(waves + 7) / 8, 256, 0, stream>>>(enc, order, We, be, enc_att);
    }

    // helper: grid for the 2x4-blocked generic GEMM (8 waves per 256-thr block)
    auto gemm_grid = [](int M, int N) {
        int waves = (M >> 5) * (((N >> 4) + 3) >> 2);
        return (waves + 7) / 8;
    };

    // 3) sequential scan
    for (int t = 0; t < TT; ++t) {
        // dec_att = h @ Wd^T + bd : 4*8 = 32 waves
        gemm_wmma_f32<<<gemm_grid(BB, AD), 256, 0, stream>>>(
            h, DD, Wd, DD, bd,
            nullptr, 0, nullptr, 0, 0, nullptr,
            dec_att, AD, BB, AD, DD, nullptr, 0);
        // scores over (b,p)
        scores_kernel<<<(BB * PP + 7) / 8, 256, 0, stream>>>(enc_att, dec_att, wf, bf, scores);
        // softmax over P
        softmax_kernel<<<BB, 256, 0, stream>>>(scores, alpha);
        // awe (ungated)
        awe_kernel<<<dim3(BB, ENCD / 256), 256, 0, stream>>>(alpha, enc, order, awe);
        // gate pre-activation = h @ Wbeta^T + bbeta : 4*32 = 128 waves
        gemm_wmma_f32<<<gemm_grid(BB, ENCD), 256, 0, stream>>>(
            h, DD, Wbeta, DD, bbeta,
            nullptr, 0, nullptr, 0, 0, nullptr,
            gatepre, ENCD, BB, ENCD, DD, nullptr, 0);
        // x = [emb_tok | sigmoid(gatepre) * awe]
        {
            long n = (long)BB * (ED + ENCD);
            xassemble_kernel<<<(int)((n + 255) / 256), 256, 0, stream>>>(
                emb, caps, order, gatepre, awe, xbuf, t);
        }
        // gates = x @ W_ih^T + b_ih + h @ W_hh^T + b_hh  (fused, K=2560 then K2=512)
        gemm_wmma_f32<<<gemm_grid(BB, 4 * DD), 256, 0, stream>>>(
            xbuf, ED + ENCD, W_ih, ED + ENCD, b_ih,
            h, DD, W_hh, DD, DD, b_hh,
            gates, 4 * DD, BB, 4 * DD, ED + ENCD, nullptr, 0);
        // LSTM cell
        lstm_kernel<<<(BB * DD + 255) / 256, 256, 0, stream>>>(gates, h, cbuf, hnew, declen, t);
        // preds = h_new @ Wfc^T + bfc, masked, straight into d_out
        // row stride = T*V, column offset = t*V. 4*157 = 628 waves.
        gemm_wmma_f32<<<gemm_grid(BB, VD), 256, 0, stream>>>(
            hnew, DD, Wfc, DD, bfc,
            nullptr, 0, nullptr, 0, 0, nullptr,
            out_preds + (long)t * VD, (long)TT * VD, BB, VD, DD, declen, t);
    }
}